// Sparsemax_59803124630069
// MI455X (gfx1250) — compile-verified
//
#include <hip/hip_runtime.h>
#include <stdint.h>

#define ROWLEN   2048
#define NTHREADS 256
#define NWAVES   (NTHREADS / 32)

typedef unsigned int u32x4 __attribute__((ext_vector_type(4)));
typedef int          i32x8 __attribute__((ext_vector_type(8)));
typedef int          i32x4 __attribute__((ext_vector_type(4)));
typedef float        f32x4 __attribute__((ext_vector_type(4)));

// Blockwide (sum, count) reduction. wave32 shuffle tree, then LDS across 8 waves.
// Result is broadcast to all threads (bitwise identical -> uniform branches OK).
__device__ __forceinline__ void block_reduce_pair(float& s, float& c, float* red) {
#pragma unroll
    for (int off = 16; off > 0; off >>= 1) {
        s += __shfl_xor(s, off, 32);
        c += __shfl_xor(c, off, 32);
    }
    const int wave = threadIdx.x >> 5;
    const int lane = threadIdx.x & 31;
    if (lane == 0) { red[2 * wave] = s; red[2 * wave + 1] = c; }
    __syncthreads();
    if (threadIdx.x == 0) {
        float S = 0.f, C = 0.f;
#pragma unroll
        for (int w = 0; w < NWAVES; ++w) { S += red[2 * w]; C += red[2 * w + 1]; }
        red[16] = S;
        red[17] = C;
    }
    __syncthreads();
    s = red[16];
    c = red[17];
    __syncthreads();  // protect `red` reuse on the next iteration
}

__global__ __launch_bounds__(NTHREADS)
void sparsemax_rows_kernel(const float* __restrict__ z, float* __restrict__ out) {
    __shared__ float row[ROWLEN];   // 8 KB staging buffer (one row)
    __shared__ float red[32];

    const int     t    = threadIdx.x;
    const size_t  base = (size_t)blockIdx.x * ROWLEN;
    const float*  zr   = z + base;
    float*        outr = out + base;

    // ---- Stage row HBM -> LDS via Tensor Data Mover --------------------------
#if __has_builtin(__builtin_amdgcn_tensor_load_to_lds) && __has_builtin(__builtin_amdgcn_s_wait_tensorcnt)
    if (t < 32) {  // one TDM op per workgroup (per-wave DMA, EXEC ignored)
        const uint32_t lds_off = (uint32_t)(uintptr_t)(&row[0]);   // ADDR[31:0] = LDS offset
        const uint64_t ga      = (uint64_t)(uintptr_t)zr;

        // D# group 0: count=1 | lds_addr | global_addr[56:0] | type=2
        u32x4 g0 = {};
        g0[0] = 1u;
        g0[1] = lds_off;
        g0[2] = (uint32_t)ga;
        g0[3] = (uint32_t)((ga >> 32) & 0x01FFFFFFu) | (2u << 30);

        // D# group 1: data_size=4B, tensor_dim0=2048, tensor_dim1=1,
        //             tile_dim0=2048, tile_dim1=1, dim0_stride=2048
        i32x8 g1 = {};
        g1[0] = (int)(2u << 16);                   // [17:16] data_size = 2 (4 bytes)
        g1[1] = (int)((uint32_t)ROWLEN << 16);     // [63:48] tensor_dim0 lo16
        g1[2] = (int)(1u << 16);                   // [95:80] tensor_dim1 = 1
        g1[3] = (int)((uint32_t)ROWLEN << 16);     // [127:112] tile_dim0
        g1[4] = 1;                                 // [143:128] tile_dim1 = 1
        g1[5] = ROWLEN;                            // [191:160] tensor_dim0_stride lo32
        g1[6] = 0;
        g1[7] = 0;

        i32x4 z4 = {0, 0, 0, 0};                   // groups 2/3 unused (<=2D tensor)
        i32x8 z8 = {0, 0, 0, 0, 0, 0, 0, 0};       // extra group (clang-23 6-arg form)
        // cpol = 1 -> TH_NT: row is streamed exactly once, keep it out of L2 LRU
        __builtin_amdgcn_tensor_load_to_lds(g0, g1, z4, z4, z8, 1);
        __builtin_amdgcn_s_wait_tensorcnt(0);      // TENSORcnt == 0 -> data in LDS
    }
    __syncthreads();
#else
    {
        const f32x4* src = (const f32x4*)zr;
        f32x4*       dst = (f32x4*)row;
        dst[t]            = src[t];
        dst[t + NTHREADS] = src[t + NTHREADS];
    }
    __syncthreads();
#endif

    // ---- Pull this thread's 8 elements into registers ------------------------
    const f32x4 a = ((const f32x4*)row)[t];
    const f32x4 b = ((const f32x4*)row)[t + NTHREADS];
    float v[8] = {a[0], a[1], a[2], a[3], b[0], b[1], b[2], b[3]};

    // ---- Michelot fixed-point iteration for tau ------------------------------
    float tau;
    {
        float s = 0.f, c = 0.f;
#pragma unroll
        for (int k = 0; k < 8; ++k) s += v[k];
        block_reduce_pair(s, c, red);
        tau = (s - 1.0f) * (1.0f / (float)ROWLEN);
    }

    for (int it = 0; it < 64; ++it) {
        float s = 0.f, c = 0.f;
#pragma unroll
        for (int k = 0; k < 8; ++k) {
            if (v[k] > tau) { s += v[k]; c += 1.0f; }
        }
        block_reduce_pair(s, c, red);
        const float nt = (s - 1.0f) / c;   // c >= 1 always (max element stays in support)
        if (!(nt > tau)) break;            // uniform: tau monotonically non-decreasing
        tau = nt;
    }

    // ---- Write out max(z - tau, 0); NT stores (output never re-read) ---------
    f32x4 o0, o1;
    o0[0] = fmaxf(v[0] - tau, 0.f);
    o0[1] = fmaxf(v[1] - tau, 0.f);
    o0[2] = fmaxf(v[2] - tau, 0.f);
    o0[3] = fmaxf(v[3] - tau, 0.f);
    o1[0] = fmaxf(v[4] - tau, 0.f);
    o1[1] = fmaxf(v[5] - tau, 0.f);
    o1[2] = fmaxf(v[6] - tau, 0.f);
    o1[3] = fmaxf(v[7] - tau, 0.f);
    __builtin_nontemporal_store(o0, ((f32x4*)outr) + t);
    __builtin_nontemporal_store(o1, ((f32x4*)outr) + t + NTHREADS);
}

extern "C" void kernel_launch(void* const* d_in, const int* in_sizes, int n_in,
                              void* d_out, int out_size, void* d_ws, size_t ws_size,
                              hipStream_t stream) {
    (void)n_in; (void)out_size; (void)d_ws; (void)ws_size;
    const float* z   = (const float*)d_in[0];
    float*       out = (float*)d_out;
    const int    nrows = in_sizes[0] / ROWLEN;   // 2*16*2048 = 65536
    sparsemax_rows_kernel<<<nrows, NTHREADS, 0, stream>>>(z, out);
}